// SSKernelDiag_28853590294578
// MI455X (gfx1250) — compile-verified
//
#include <hip/hip_runtime.h>

// S4D Vandermonde kernel for MI455X (gfx1250, wave32).
//
// out[h, l] = 2*Re( sum_n Cm[h,n] * exp(dtA[h,n] * l) ),  l in [0, L)
//   dtA = (-exp(log_w_real) + i*w_imag) * exp(log_dt[h])
//   Cm  = (C_r + i C_i) * (exp(dtA) - 1) / w
//
// GEMM formulation (l = 256*g + 16*c + a, c,a in [0,16)):
//   Out_g(16x16) = P_g(16 x 2N) @ Q(2N x 16),  K = 2N = 128
//   P_g[c, 2n+{0,1}] = Re/Im( 2*Cm_n * exp(dtA_n*16c) * exp(dtA_n*256g) )
//   Q[2n+0, a] = Re exp(dtA_n * a);  Q[2n+1, a] = -Im exp(dtA_n * a)
// 32 steps of V_WMMA_F32_16X16X4_F32 (exact f32 GEMM).
//
// Block = one h (256 threads, 8 waves); wave g produces taps [256g, 256g+256).
// Phase 1 precomputes all transcendentals into LDS tables; the main loop is
// pure DS-load + complex-multiply + WMMA.

typedef __attribute__((ext_vector_type(2))) float v2f;
typedef __attribute__((ext_vector_type(8))) float v8f;

#define NMAX 64   // modes per head
#define GMAX 8    // 256-tap groups per head (L/256)

__global__ __launch_bounds__(256) void s4d_vandermonde_wmma(
    const float* __restrict__ C,           // (1, H, N, 2)
    const float* __restrict__ log_dt,      // (H,)
    const float* __restrict__ log_w_real,  // (H, N)
    const float* __restrict__ w_imag,      // (H, N)
    float* __restrict__ out,               // (1, H, L)
    int H, int N, int L)
{
  __shared__ float2 sDtA[NMAX];        // (ar, ai)
  __shared__ float2 sCm[NMAX];         // 2*Cm
  __shared__ float2 sQ[NMAX][16];      // B frags: (Qr, -Qi) = exp(dtA*a)
  __shared__ float2 sZC[NMAX][16];     // 2*Cm * exp(dtA*16c)
  __shared__ float2 sR[NMAX][GMAX];    // exp(dtA*256g)

  const int h    = blockIdx.x;
  const int tid  = threadIdx.x;
  const int lane = tid & 31;
  const int g    = tid >> 5;           // this wave's l-group
  const int hilo = lane >> 4;          // half-wave select
  const int m16  = lane & 15;
  const int nthreads = blockDim.x;

  // ---- Phase 1a: per-mode parameters dtA, 2*Cm -------------------------
  const float dt = expf(log_dt[h]);
  if (tid < N) {
    const int n  = tid;
    const int hn = h * N + n;
    const float wr = -expf(log_w_real[hn]);
    const float wi = w_imag[hn];
    const float ar = wr * dt;
    const float ai = wi * dt;

    // 2 * C * (exp(dtA) - 1) / w
    float s, c;
    sincosf(ai, &s, &c);
    const float er  = expf(ar);
    const float exr = er * c - 1.0f;
    const float exi = er * s;
    const float inv = 1.0f / (wr * wr + wi * wi);
    const float fr  = (exr * wr + exi * wi) * inv;
    const float fi  = (exi * wr - exr * wi) * inv;
    const float c0r = C[2 * hn + 0];
    const float c0i = C[2 * hn + 1];

    sDtA[n] = make_float2(ar, ai);
    sCm[n]  = make_float2(2.0f * (c0r * fr - c0i * fi),
                          2.0f * (c0r * fi + c0i * fr));
  }
  __syncthreads();

  // ---- Phase 1b: transcendental tables ---------------------------------
  // sQ[n][a] = (Re, -Im) exp(dtA*a);  sZC[n][c] = 2Cm * exp(dtA*16c)
  for (int idx = tid; idx < N * 16; idx += nthreads) {
    const int n = idx >> 4;
    const int p = idx & 15;
    const float ar = sDtA[n].x, ai = sDtA[n].y;

    float s, c;
    // Q table (a = p)
    const float eq = expf(ar * (float)p);
    sincosf(ai * (float)p, &s, &c);
    sQ[n][p] = make_float2(eq * c, -eq * s);

    // ZC table (l = 16*p), with 2*Cm folded in
    const float pl = (float)(16 * p);
    const float ez = expf(ar * pl);
    sincosf(ai * pl, &s, &c);
    const float zr = ez * c, zi = ez * s;
    const float cmr = sCm[n].x, cmi = sCm[n].y;
    sZC[n][p] = make_float2(cmr * zr - cmi * zi, cmr * zi + cmi * zr);
  }
  for (int idx = tid; idx < N * GMAX; idx += nthreads) {
    const int n  = idx >> 3;
    const int gg = idx & (GMAX - 1);
    const float ar = sDtA[n].x, ai = sDtA[n].y;
    const float pl = (float)(256 * gg);
    const float er = expf(ar * pl);
    float s, c;
    sincosf(ai * pl, &s, &c);
    sR[n][gg] = make_float2(er * c, er * s);
  }
  __syncthreads();

  // ---- Phase 2: K-loop of f32 WMMAs ------------------------------------
  // A(16x4) layout: lane = M row (chunk c), half-wave selects K pair
  // {2*hilo, 2*hilo+1}; vgpr0 = even k (real row), vgpr1 = odd k (imag row).
  // B mirrored: lane = N col (tap a), half-wave selects K row pair.
  v8f acc = {};
  const int tsteps = N >> 1;           // K = 2N, K-step 4
#pragma unroll 4
  for (int t = 0; t < tsteps; ++t) {
    const int n = 2 * t + hilo;        // this lane's mode for this K-step

    const float2 zc = sZC[n][m16];     // 2Cm * exp(dtA*16c)
    const float2 r  = sR[n][g];        // exp(dtA*256g)  (broadcast per half)
    const float2 q  = sQ[n][m16];      // (Qr, -Qi)

    v2f afrag;                         // P[c, 2n], P[c, 2n+1]
    afrag.x = zc.x * r.x - zc.y * r.y;
    afrag.y = zc.x * r.y + zc.y * r.x;
    v2f bfrag;
    bfrag.x = q.x;
    bfrag.y = q.y;

    acc = __builtin_amdgcn_wmma_f32_16x16x4_f32(
        /*neg_a=*/false, afrag, /*neg_b=*/false, bfrag,
        /*c_mod=*/(short)0, acc, /*reuse_a=*/false, /*reuse_b=*/false);
  }

  // D layout: vgpr j -> M = j + 8*hilo (chunk c), N = lane&15 (tap a).
  float* op = out + (size_t)h * L + 256 * g;
  const int mbase = 8 * hilo;
#pragma unroll
  for (int j = 0; j < 8; ++j) {
    op[16 * (mbase + j) + m16] = acc[j];
  }
}

extern "C" void kernel_launch(void* const* d_in, const int* in_sizes, int n_in,
                              void* d_out, int out_size, void* d_ws, size_t ws_size,
                              hipStream_t stream) {
  const float* C          = (const float*)d_in[0];  // (1,H,N,2)
  const float* log_dt     = (const float*)d_in[1];  // (H,)
  const float* log_w_real = (const float*)d_in[2];  // (H,N)
  const float* w_imag     = (const float*)d_in[3];  // (H,N)
  float*       out        = (float*)d_out;          // (1,H,L)

  const int H = in_sizes[1];
  const int N = in_sizes[2] / H;       // 64
  const int L = out_size / H;          // 2048 (CCH == 1)

  const int waves_per_h = L / 256;     // 8 waves, one per 256-tap group
  dim3 grid(H);
  dim3 block(32 * waves_per_h);        // 256 threads

  s4d_vandermonde_wmma<<<grid, block, 0, stream>>>(
      C, log_dt, log_w_real, w_imag, out, H, N, L);
}